// LGSSM_42649025249669
// MI455X (gfx1250) — compile-verified
//
#include <hip/hip_runtime.h>
#include <math.h>

// ---------------------------------------------------------------------------
// LGSSM Kalman filter, restructured for MI455X (gfx1250, wave32, WMMA):
//   K1: data-independent Riccati chain (chunked + warm-up, 64 waves)
//       -> K_t (5x32), P_t (5x5), logdet S_t
//   K2: per-sequence affine state scan + log-lik, 16 sequences per wave,
//       obs-dim algebra on V_WMMA_F32_16X16X4_F32 (chunked + warm-up)
//   K3: broadcast covs (pure bandwidth)
//   K4: reduce per-chunk log-lik partials
// ---------------------------------------------------------------------------

#define STATE   5
#define OBS     32
#define TLEN    8192
#define BATCH   256
#define CHUNK   128
#define NCHUNK  (TLEN / CHUNK)   // 64
#define WARM    128
#define SEQPW   16               // sequences per wave in K2
#define LOG2PI  1.8378770664093454f

typedef __attribute__((ext_vector_type(2))) float v2f;
typedef __attribute__((ext_vector_type(8))) float v8f;

__device__ __forceinline__ v8f wmma16x16x4(v2f a, v2f b, v8f c) {
#if defined(__HIP_DEVICE_COMPILE__)
  // D = A(16x4 f32) * B(4x16 f32) + C(16x16 f32)
  return __builtin_amdgcn_wmma_f32_16x16x4_f32(false, a, false, b, (short)0, c,
                                               false, false);
#else
  (void)a; (void)b;
  return c;
#endif
}

// 5x5 SPD Cholesky -> inverse + logdet, fully unrolled, per-lane.
__device__ __forceinline__ void chol_inv5(const float P[25], float Pi[25],
                                          float& ld) {
  float L[15];            // lower incl diag, packed idx i*(i+1)/2+j
  float prod = 1.f;
#pragma unroll
  for (int j = 0; j < 5; ++j) {
    float d = P[j * 5 + j];
#pragma unroll
    for (int k = 0; k < j; ++k) d -= L[j*(j+1)/2 + k] * L[j*(j+1)/2 + k];
    float sdg = sqrtf(d);
    float inv = 1.f / sdg;
    L[j*(j+1)/2 + j] = sdg;
    prod *= sdg;
#pragma unroll
    for (int i = j + 1; i < 5; ++i) {
      float a = P[i * 5 + j];
#pragma unroll
      for (int k = 0; k < j; ++k) a -= L[i*(i+1)/2 + k] * L[j*(j+1)/2 + k];
      L[i*(i+1)/2 + j] = a * inv;
    }
  }
  ld = 2.f * logf(prod);
  float Li[15];
#pragma unroll
  for (int j = 0; j < 5; ++j) {
    Li[j*(j+1)/2 + j] = 1.f / L[j*(j+1)/2 + j];
#pragma unroll
    for (int i = j + 1; i < 5; ++i) {
      float a = 0.f;
#pragma unroll
      for (int k = j; k < i; ++k) a += L[i*(i+1)/2 + k] * Li[k*(k+1)/2 + j];
      Li[i*(i+1)/2 + j] = -a / L[i*(i+1)/2 + i];
    }
  }
#pragma unroll
  for (int a = 0; a < 5; ++a)
#pragma unroll
    for (int b = a; b < 5; ++b) {
      float s2 = 0.f;
      const int lo = (a > b) ? a : b;
#pragma unroll
      for (int k = 0; k < 5; ++k)
        if (k >= lo) s2 += Li[k*(k+1)/2 + a] * Li[k*(k+1)/2 + b];
      Pi[a * 5 + b] = s2;
      Pi[b * 5 + a] = s2;
    }
}

// ---------------------------------------------------------------------------
// K1: Riccati precompute. One wave per 128-step chunk; warm-up from P=I
// converges geometrically (contraction ~0.35/step -> 128 steps >> fp32 eps).
// Lanes: redundant 5x5 chain; lane o owns column o of K_t (o = obs index).
// ---------------------------------------------------------------------------
__global__ void __launch_bounds__(32)
k1_riccati(const float* __restrict__ A, const float* __restrict__ C,
           const float* __restrict__ Qlog, const float* __restrict__ Rlog,
           float* __restrict__ Kt, float* __restrict__ Pout,
           float* __restrict__ ldS) {
  const int lane   = threadIdx.x;
  const int chunk  = blockIdx.x;
  const int tstart = chunk * CHUNK;
  const int warm   = (chunk == 0) ? 0 : WARM;

  float Am[25];
#pragma unroll
  for (int i = 0; i < 25; ++i) Am[i] = A[i];
  float Qd[5];
#pragma unroll
  for (int i = 0; i < 5; ++i) Qd[i] = expf(Qlog[i]);
  float cL[5];
#pragma unroll
  for (int r = 0; r < 5; ++r) cL[r] = C[lane * 5 + r];
  const float rinvL = expf(-Rlog[lane]);
  float ldR = 0.f;
#pragma unroll
  for (int o = 0; o < OBS; ++o) ldR += Rlog[o];   // logdet(diag(exp(Rlog)))

  // G = C^T R^-1 C, cooperative outer products + shuffle reduce.
  float G[25];
#pragma unroll
  for (int a = 0; a < 5; ++a)
#pragma unroll
    for (int b = a; b < 5; ++b) {
      float g = rinvL * cL[a] * cL[b];
#pragma unroll
      for (int off = 16; off > 0; off >>= 1) g += __shfl_xor(g, off, 32);
      G[a * 5 + b] = g;
      G[b * 5 + a] = g;
    }

  float P[25];
#pragma unroll
  for (int i = 0; i < 25; ++i) P[i] = 0.f;
#pragma unroll
  for (int i = 0; i < 5; ++i) P[i * 5 + i] = 1.f;   // P0 = I

  for (int t = tstart - warm; t < tstart + CHUNK; ++t) {
    // P_pred = A P A^T + diag(Qd)
    float T1[25], Pp[25];
#pragma unroll
    for (int i = 0; i < 5; ++i)
#pragma unroll
      for (int j = 0; j < 5; ++j) {
        float s = 0.f;
#pragma unroll
        for (int k = 0; k < 5; ++k) s += Am[i * 5 + k] * P[k * 5 + j];
        T1[i * 5 + j] = s;
      }
#pragma unroll
    for (int i = 0; i < 5; ++i)
#pragma unroll
      for (int j = 0; j < 5; ++j) {
        float s = 0.f;
#pragma unroll
        for (int k = 0; k < 5; ++k) s += T1[i * 5 + k] * Am[j * 5 + k];
        Pp[i * 5 + j] = s + ((i == j) ? Qd[i] : 0.f);
      }
    float Pinv[25], ldP;
    chol_inv5(Pp, Pinv, ldP);
    float M[25];
#pragma unroll
    for (int i = 0; i < 25; ++i) M[i] = Pinv[i] + G[i];
    float Pn[25], ldM;
    chol_inv5(M, Pn, ldM);    // P_new = (P_pred^-1 + G)^-1

    if (t >= tstart) {
      // K_t[r][o] = rinv_o * sum_s P_new[r][s] C[o][s]   (this lane = obs o)
#pragma unroll
      for (int r = 0; r < 5; ++r) {
        float kv = 0.f;
#pragma unroll
        for (int s = 0; s < 5; ++s) kv += Pn[r * 5 + s] * cL[s];
        Kt[t * (5 * OBS) + r * OBS + lane] = kv * rinvL;
      }
      if (lane == 0) {
        ldS[t] = ldR + ldP + ldM;   // logdet S = logdetR+logdetPpred+logdetM
#pragma unroll
        for (int i = 0; i < 25; ++i) Pout[t * 25 + i] = Pn[i];
      }
    }
#pragma unroll
    for (int i = 0; i < 25; ++i) P[i] = Pn[i];
  }
}

// ---------------------------------------------------------------------------
// K2: state scan + log-lik. 16 sequences per wave, WMMA f32 16x16x4.
//   E  = C * z_pred^T - Y^T          (= -V^T)      4 WMMAs (2 obs tiles, K=5)
//   D2 = K_t * E (= -K v),  U = C^T R^-1 * E (= -u)   8+8 WMMAs (K=32)
//   z_new = z_pred - bcast(D2);  ll terms: v^T R^-1 v and u . (Kv)
// ---------------------------------------------------------------------------
__global__ void __launch_bounds__(32)
k2_filter(const float* __restrict__ y, const float* __restrict__ A,
          const float* __restrict__ C, const float* __restrict__ Rlog,
          const float* __restrict__ Kt, const float* __restrict__ ldS,
          float* __restrict__ states, float* __restrict__ llpart) {
  const int lane   = threadIdx.x;
  const int s      = lane & 15;    // sequence slot in this wave
  const int hi     = lane >> 4;    // wave half
  const int chunk  = blockIdx.x % NCHUNK;
  const int grp    = blockIdx.x / NCHUNK;
  const int b      = grp * SEQPW + s;
  const int tstart = chunk * CHUNK;
  const int warm   = (chunk == 0) ? 0 : WARM;

  float Am[25];
#pragma unroll
  for (int i = 0; i < 25; ++i) Am[i] = A[i];

  // C as WMMA-A operand: tiles n (obs rows), ksteps over state K (pad 5->8).
  v2f cA[2][2];
#pragma unroll
  for (int n = 0; n < 2; ++n)
#pragma unroll
    for (int ks = 0; ks < 2; ++ks) {
      const int row = n * 16 + s;
      const int k0  = ks * 4 + 2 * hi;
      cA[n][ks].x = (k0     < 5) ? C[row * 5 + k0]     : 0.f;
      cA[n][ks].y = (k0 + 1 < 5) ? C[row * 5 + k0 + 1] : 0.f;
    }
  // C^T R^-1 as WMMA-A operand (state rows, K = obs, 8 ksteps).
  v2f uA[8];
#pragma unroll
  for (int ks = 0; ks < 8; ++ks) {
    const int k0 = ks * 4 + 2 * hi;
    const float r0 = expf(-Rlog[k0]);
    const float r1 = expf(-Rlog[k0 + 1]);
    uA[ks].x = (s < 5) ? C[k0 * 5 + s] * r0       : 0.f;
    uA[ks].y = (s < 5) ? C[(k0 + 1) * 5 + s] * r1 : 0.f;
  }
  // rinv for the obs this lane holds in E layout: obs = n*16 + j + 8*hi
  float rl[16];
#pragma unroll
  for (int n = 0; n < 2; ++n)
#pragma unroll
    for (int j = 0; j < 8; ++j) rl[n * 8 + j] = expf(-Rlog[n * 16 + j + 8 * hi]);

  float z[5];
#pragma unroll
  for (int r = 0; r < 5; ++r) z[r] = 0.f;
  float acc1 = 0.f, acc2 = 0.f, accL = 0.f;

  for (int t = tstart - warm; t < tstart + CHUNK; ++t) {
    const bool active = (t >= tstart);
    if (t + 8 < TLEN)
      __builtin_prefetch(&y[(size_t)(b * TLEN + t + 8) * OBS], 0, 0);

    // z_pred = A z (redundant per lane; each lane holds its sequence's z)
    float zp[5];
#pragma unroll
    for (int r = 0; r < 5; ++r) {
      float a = 0.f;
#pragma unroll
      for (int c2 = 0; c2 < 5; ++c2) a += Am[r * 5 + c2] * z[c2];
      zp[r] = a;
    }
    // B operand from z_pred^T (K = state)
    v2f bz0, bz1;
    bz0.x = hi ? zp[2] : zp[0];
    bz0.y = hi ? zp[3] : zp[1];
    bz1.x = hi ? 0.f   : zp[4];
    bz1.y = 0.f;

    // C-accumulator = -Y^T in D layout (VGPR j <-> obs row j + 8*hi)
    const float* yrow = &y[(size_t)(b * TLEN + t) * OBS];
    v8f e0, e1;
    {
      const float4 ya = *(const float4*)(yrow + 8 * hi);
      const float4 yb = *(const float4*)(yrow + 8 * hi + 4);
      const float4 yc = *(const float4*)(yrow + 16 + 8 * hi);
      const float4 yd = *(const float4*)(yrow + 16 + 8 * hi + 4);
      e0[0] = -ya.x; e0[1] = -ya.y; e0[2] = -ya.z; e0[3] = -ya.w;
      e0[4] = -yb.x; e0[5] = -yb.y; e0[6] = -yb.z; e0[7] = -yb.w;
      e1[0] = -yc.x; e1[1] = -yc.y; e1[2] = -yc.z; e1[3] = -yc.w;
      e1[4] = -yd.x; e1[5] = -yd.y; e1[6] = -yd.z; e1[7] = -yd.w;
    }
    // E = C * z_pred^T - Y^T   (= -V^T)
    e0 = wmma16x16x4(cA[0][0], bz0, e0);
    e0 = wmma16x16x4(cA[0][1], bz1, e0);
    e1 = wmma16x16x4(cA[1][0], bz0, e1);
    e1 = wmma16x16x4(cA[1][1], bz1, e1);

    if (active) {
#pragma unroll
      for (int j = 0; j < 8; ++j) {
        acc1 += rl[j] * e0[j] * e0[j];        // v^T R^-1 v (half per lane)
        acc1 += rl[8 + j] * e1[j] * e1[j];
      }
      accL += ldS[t];
    }

    // D2 = K_t * E, U = (C^T R^-1) * E, shared B operand built from E.
    v8f D2 = {};
    v8f U  = {};
#pragma unroll
    for (int ks = 0; ks < 8; ++ks) {
      const int k0 = ks * 4 + 2 * hi;
      v2f kA;
      kA.x = (s < 5) ? Kt[t * (5 * OBS) + s * OBS + k0]     : 0.f;
      kA.y = (s < 5) ? Kt[t * (5 * OBS) + s * OBS + k0 + 1] : 0.f;
      v2f bv;
#pragma unroll
      for (int p = 0; p < 2; ++p) {
        const int oL = ks * 4 + p;          // obs needed by lo lanes
        const int mm = oL & 15;             // oL and oL+2 share a wave half
        const int n  = oL >> 4;
        const int srcBase = (mm >= 8) ? 16 : 0;
        const float vLoR = (n == 0) ? e0[mm & 7] : e1[mm & 7];
        const float vHiR = (n == 0) ? e0[(mm & 7) + 2] : e1[(mm & 7) + 2];
        const float lov = __shfl(vLoR, s + srcBase, 32);
        const float hiv = __shfl(vHiR, s + srcBase, 32);
        const float bvp = hi ? hiv : lov;
        if (p == 0) bv.x = bvp; else bv.y = bvp;
      }
      D2 = wmma16x16x4(kA, bv, D2);
      U  = wmma16x16x4(uA[ks], bv, U);
    }

    // Broadcast d = -K v (rows 0..4, lo lanes of D2), update state.
    float d0 = __shfl(D2[0], s, 32), d1 = __shfl(D2[1], s, 32),
          d2 = __shfl(D2[2], s, 32), d3 = __shfl(D2[3], s, 32),
          d4 = __shfl(D2[4], s, 32);
    z[0] = zp[0] - d0; z[1] = zp[1] - d1; z[2] = zp[2] - d2;
    z[3] = zp[3] - d3; z[4] = zp[4] - d4;

    if (active) {
      const float u0 = __shfl(U[0], s, 32), u1 = __shfl(U[1], s, 32),
                  u2 = __shfl(U[2], s, 32), u3 = __shfl(U[3], s, 32),
                  u4 = __shfl(U[4], s, 32);
      acc2 += u0 * d0 + u1 * d1 + u2 * d2 + u3 * d3 + u4 * d4;  // u^T Pn u
      if (hi == 0) {
        const size_t so = ((size_t)b * TLEN + t) * STATE;
        states[so + 0] = z[0]; states[so + 1] = z[1]; states[so + 2] = z[2];
        states[so + 3] = z[3]; states[so + 4] = z[4];
      }
    }
  }
  // combine the two obs halves of acc1; acc2/accL identical across halves
  const float acc1full = acc1 + __shfl_xor(acc1, 16, 32);
  if (hi == 0)
    llpart[(size_t)b * NCHUNK + chunk] = acc1full + accL - acc2;
}

// ---------------------------------------------------------------------------
// K3: covs broadcast (covs[b][t][i][j] = P_t[i][j]); pure bandwidth.
// ---------------------------------------------------------------------------
__global__ void k3_covs(const float* __restrict__ Pout,
                        float* __restrict__ covs) {
  const int rem = blockIdx.x * blockDim.x + threadIdx.x;   // < TLEN*25
  const int b   = blockIdx.y;
  if (rem < TLEN * 25)
    covs[(size_t)b * (TLEN * 25) + rem] = Pout[rem];
}

// K4: finalize per-sequence log-likelihood.
__global__ void k4_ll(const float* __restrict__ llpart, float* __restrict__ ll) {
  const int b = blockIdx.x * blockDim.x + threadIdx.x;
  if (b < BATCH) {
    float acc = 0.f;
    for (int c = 0; c < NCHUNK; ++c) acc += llpart[(size_t)b * NCHUNK + c];
    ll[b] = -0.5f * ((float)TLEN * (float)OBS * LOG2PI + acc);
  }
}

extern "C" void kernel_launch(void* const* d_in, const int* in_sizes, int n_in,
                              void* d_out, int out_size, void* d_ws,
                              size_t ws_size, hipStream_t stream) {
  const float* y    = (const float*)d_in[0];
  const float* A    = (const float*)d_in[1];
  const float* C    = (const float*)d_in[2];
  const float* Qlog = (const float*)d_in[3];
  const float* Rlog = (const float*)d_in[4];

  float* states = (float*)d_out;
  float* covs   = states + (size_t)BATCH * TLEN * STATE;
  float* ll     = covs + (size_t)BATCH * TLEN * STATE * STATE;

  float* ws     = (float*)d_ws;
  float* Kt     = ws;                              // TLEN * 5 * 32
  float* Pout   = Kt + (size_t)TLEN * 5 * OBS;     // TLEN * 25
  float* ldS    = Pout + (size_t)TLEN * 25;        // TLEN
  float* llpart = ldS + TLEN;                      // BATCH * NCHUNK

  k1_riccati<<<NCHUNK, 32, 0, stream>>>(A, C, Qlog, Rlog, Kt, Pout, ldS);
  k2_filter<<<(BATCH / SEQPW) * NCHUNK, 32, 0, stream>>>(y, A, C, Rlog, Kt,
                                                         ldS, states, llpart);
  k3_covs<<<dim3((TLEN * 25 + 255) / 256, BATCH), 256, 0, stream>>>(Pout, covs);
  k4_ll<<<1, 256, 0, stream>>>(llpart, ll);
}